// ElementwiseTensorProduct_46540265620256
// MI455X (gfx1250) — compile-verified
//
#include <hip/hip_runtime.h>
#include <stdint.h>

// ---------------------------------------------------------------------------
// ElementwiseTensorProduct  (e3nn-style), MI455X / gfx1250
//
// Segments (l1,l2): (0,1),(1,1),(2,1), MUL=128, BATCH=32768.
// Memory-bound (~3.4 FLOP/B vs 23.3 TB/s HBM => ~32us floor): WMMA cannot
// help; the CDNA5 path that matters is TDM (tensor_load_to_lds) streaming of
// the input rows into LDS + conflict-free LDS broadcast of the CG table.
// ---------------------------------------------------------------------------

#define MULS    128
#define ROWLEN  1152          // floats per input row (both f1 and f2)
#define OUTLEN  3456          // floats per output row
#define RPB     4             // batch rows per block (== 8 waves/2)
#define NTHREADS 256

// CG table layout in workspace / LDS (float offsets)
#define CG_0_11 0             // (l=0,1,1): 1*3*3
#define CG_1_11 9             // (l=1,1,1): 3*3*3
#define CG_2_11 36            // (l=2,1,1): 5*3*3
#define CG_1_21 81            // (l=1,2,1): 3*5*3
#define CG_2_21 126           // (l=2,2,1): 5*5*3
#define CG_3_21 201           // (l=3,2,1): 7*5*3
#define CG_TOTAL 306

typedef unsigned int u32;
typedef unsigned int v4u __attribute__((ext_vector_type(4)));
typedef int          v8i __attribute__((ext_vector_type(8)));
typedef int          v4i __attribute__((ext_vector_type(4)));

// ============================ main kernel ==================================
// Order: issue TDM (async) -> copy CG table while TDM is in flight ->
//        s_wait_tensorcnt 0 -> barrier -> FMA core.

__global__ __launch_bounds__(NTHREADS)
void etp_kernel(const float* __restrict__ f1, const float* __restrict__ f2,
                const float* __restrict__ cgws, float* __restrict__ out) {
  __shared__ float shF1[RPB * ROWLEN];
  __shared__ float shF2[RPB * ROWLEN];
  __shared__ float shC[CG_TOTAL];

  const int t = threadIdx.x;

#if defined(__gfx1250__) && __has_builtin(__builtin_amdgcn_tensor_load_to_lds)
  // Each of the 8 waves issues exactly one TDM descriptor (EXEC-independent):
  // wave w<4: f1 row (block*4 + w) -> shF1[w];  wave w>=4: f2 row -> shF2[w-4]
  {
    const int wave = t >> 5;
    const int rl   = wave & 3;
    const float* src   = (wave < 4) ? f1 : f2;
    float*       shdst = (wave < 4) ? shF1 : shF2;
    const uint64_t ga =
        (uint64_t)(uintptr_t)(src + (size_t)(blockIdx.x * RPB + rl) * ROWLEN);
    const u32 lds = (u32)(uintptr_t)(shdst + rl * ROWLEN); // LDS byte offset

    v4u g0;
    g0.x = 1u;                                     // count=1, user descriptor
    g0.y = lds;                                    // lds_addr[31:0]
    g0.z = (u32)(ga & 0xFFFFFFFFu);                // global_addr[31:0]
    g0.w = (u32)((ga >> 32) & 0x1FFFFFFu)          // global_addr[56:32]
         | (2u << 30);                             // type = 2 ("image")

    v8i g1;
    g1[0] = (int)(2u << 16);                       // wg_mask=0, data_size=4B
    g1[1] = (int)((u32)ROWLEN << 16);              // tensor_dim0[15:0]
    g1[2] = (int)(1u << 16);                       // dim0[31:16]=0, tensor_dim1=1
    g1[3] = (int)((u32)ROWLEN << 16);              // dim1 hi=0, tile_dim0=1152
    g1[4] = 1;                                     // tile_dim1=1, tile_dim2=0
    g1[5] = ROWLEN;                                // tensor_dim0_stride lo32
    g1[6] = (int)((u32)ROWLEN << 16);              // stride0 hi=0, stride1 lo16
    g1[7] = 0;                                     // stride1 hi
    v4i gz4 = {0, 0, 0, 0};                        // groups 2/3 unused (<=2D)
    v8i gz8 = {0, 0, 0, 0, 0, 0, 0, 0};            // extra group (6-arg form)

    __builtin_amdgcn_tensor_load_to_lds(g0, g1, gz4, gz4, gz8, 0);
  }

  // CG table -> LDS while the TDM transfer is in flight (306 floats)
  for (int i = t; i < CG_TOTAL; i += NTHREADS) shC[i] = cgws[i];

  __builtin_amdgcn_s_wait_tensorcnt(0);
#else
  // Fallback: cooperative coalesced copy (rows are contiguous)
  for (int i = t; i < CG_TOTAL; i += NTHREADS) shC[i] = cgws[i];
  {
    const float* s1 = f1 + (size_t)blockIdx.x * RPB * ROWLEN;
    const float* s2 = f2 + (size_t)blockIdx.x * RPB * ROWLEN;
    for (int i = t; i < RPB * ROWLEN; i += NTHREADS) {
      shF1[i] = s1[i];
      shF2[i] = s2[i];
    }
  }
#endif
  __syncthreads();

  const int row_g_base = blockIdx.x * RPB;

  // Rolled on purpose: keeps VGPR pressure low (-> more resident waves for
  // store/load latency hiding); each iteration already has 27 independent
  // dot products of ILP.
#pragma unroll 1
  for (int pp = 0; pp < 2; ++pp) {
    const int pair = t + pp * NTHREADS;       // 0..511
    const int rl = pair >> 7;                 // row in block
    const int m  = pair & (MULS - 1);         // mul index
    const float* F1 = shF1 + rl * ROWLEN;
    const float* F2 = shF2 + rl * ROWLEN;
    float* o = out + (size_t)(row_g_base + rl) * OUTLEN;

    // f1: scalar (l=0), vector p (l=1), rank-2 d (l=2)
    const float s0 = F1[m];
    float p[3], d[5], a[3], b[3], c[3];
#pragma unroll
    for (int j = 0; j < 3; ++j) p[j] = F1[128 + 3*m + j];
#pragma unroll
    for (int j = 0; j < 5; ++j) d[j] = F1[512 + 5*m + j];
    // f2: three l=1 chunks
#pragma unroll
    for (int k = 0; k < 3; ++k) a[k] = F2[        3*m + k];
#pragma unroll
    for (int k = 0; k < 3; ++k) b[k] = F2[ 384 + 3*m + k];
#pragma unroll
    for (int k = 0; k < 3; ++k) c[k] = F2[ 768 + 3*m + k];

    // ---- segment 0: (0,1) scalar * vector ----
#pragma unroll
    for (int k = 0; k < 3; ++k) o[3*m + k] = s0 * a[k];

    // outer products reused by all l-outputs of a segment
    float op1[9];
#pragma unroll
    for (int j = 0; j < 3; ++j)
#pragma unroll
      for (int k = 0; k < 3; ++k) op1[j*3 + k] = p[j] * b[k];
    float op2[15];
#pragma unroll
    for (int j = 0; j < 5; ++j)
#pragma unroll
      for (int k = 0; k < 3; ++k) op2[j*3 + k] = d[j] * c[k];

    // ---- segment 1: (1,1) -> l = 0,1,2 ----
    {
      const float* C = shC + CG_0_11;
      float acc = 0.f;
#pragma unroll
      for (int q = 0; q < 9; ++q) acc = fmaf(C[q], op1[q], acc);
      o[384 + m] = acc;
    }
    {
      const float* C = shC + CG_1_11;
#pragma unroll
      for (int i = 0; i < 3; ++i) {
        float acc = 0.f;
#pragma unroll
        for (int q = 0; q < 9; ++q) acc = fmaf(C[i*9 + q], op1[q], acc);
        o[512 + 3*m + i] = acc;
      }
    }
    {
      const float* C = shC + CG_2_11;
#pragma unroll
      for (int i = 0; i < 5; ++i) {
        float acc = 0.f;
#pragma unroll
        for (int q = 0; q < 9; ++q) acc = fmaf(C[i*9 + q], op1[q], acc);
        o[896 + 5*m + i] = acc;
      }
    }

    // ---- segment 2: (2,1) -> l = 1,2,3 ----
    {
      const float* C = shC + CG_1_21;
#pragma unroll
      for (int i = 0; i < 3; ++i) {
        float acc = 0.f;
#pragma unroll
        for (int q = 0; q < 15; ++q) acc = fmaf(C[i*15 + q], op2[q], acc);
        o[1536 + 3*m + i] = acc;
      }
    }
    {
      const float* C = shC + CG_2_21;
#pragma unroll
      for (int i = 0; i < 5; ++i) {
        float acc = 0.f;
#pragma unroll
        for (int q = 0; q < 15; ++q) acc = fmaf(C[i*15 + q], op2[q], acc);
        o[1920 + 5*m + i] = acc;
      }
    }
    {
      const float* C = shC + CG_3_21;
#pragma unroll
      for (int i = 0; i < 7; ++i) {
        float acc = 0.f;
#pragma unroll
        for (int q = 0; q < 15; ++q) acc = fmaf(C[i*15 + q], op2[q], acc);
        o[2560 + 7*m + i] = acc;
      }
    }
  }
}

// ======================= CG coefficient generation =========================
// Exact port of the reference Racah formula + real<->complex SH basis change,
// evaluated in double on-device (one thread per (l_out,l1,l2) tensor).
// Runs once per launch (6 active threads), cost is negligible.

__device__ __forceinline__ double dfac(int n) {
  const double f[8] = {1., 1., 2., 6., 24., 120., 720., 5040.};
  return f[n];
}
__device__ __forceinline__ int imax2(int a, int b) { return a > b ? a : b; }
__device__ __forceinline__ int imin2(int a, int b) { return a < b ? a : b; }

struct cplx { double re, im; };
__device__ __forceinline__ cplx cmul(cplx a, cplx b) {
  cplx r; r.re = a.re*b.re - a.im*b.im; r.im = a.re*b.im + a.im*b.re; return r;
}

__device__ double su2_elem(int j1, int j2, int j3, int m1, int m2) {
  int m3 = m1 + m2;
  if (m3 < -j3 || m3 > j3) return 0.0;
  int vmin = imax2(imax2(-j1 + j2 + m3, -j1 + m1), 0);
  int vmax = imin2(imin2(j2 + j3 + m1, j3 - j1 + j2), j3 + m3);
  double C = sqrt((2.0*j3 + 1.0) *
      (dfac(j3 + j1 - j2) * dfac(j3 - j1 + j2) * dfac(j1 + j2 - j3) *
       dfac(j3 + m3) * dfac(j3 - m3)) /
      (dfac(j1 + j2 + j3 + 1) * dfac(j1 - m1) * dfac(j1 + m1) *
       dfac(j2 - m2) * dfac(j2 + m2)));
  double S = 0.0;
  for (int v = vmin; v <= vmax; ++v) {
    double sgn = ((v + j2 + m2) & 1) ? -1.0 : 1.0;
    S += sgn * (dfac(j2 + j3 + m1 - v) * dfac(j1 - m1 + v)) /
         (dfac(v) * dfac(j3 - j1 + j2 - v) * dfac(j3 + m3 - v) *
          dfac(v + j1 - j2 - m3));
  }
  return C * S;
}

__device__ void build_q(int l, cplx q[7][7]) {
  for (int i = 0; i < 7; ++i)
    for (int j = 0; j < 7; ++j) { q[i][j].re = 0.0; q[i][j].im = 0.0; }
  const double s = 0.70710678118654752440;
  for (int m = -l; m < 0; ++m) {
    q[l + m][l - m].re = s;          // 1/sqrt(2)
    q[l + m][l + m].im = -s;         // -i/sqrt(2)
  }
  q[l][l].re = 1.0;
  for (int m = 1; m <= l; ++m) {
    double sg = (m & 1) ? -1.0 : 1.0;
    q[l + m][l + m].re = sg * s;     // (-1)^m/sqrt(2)
    q[l + m][l - m].im = sg * s;     // i(-1)^m/sqrt(2)
  }
  cplx ph;                            // (-i)^l
  switch (l & 3) {
    case 0:  ph.re = 1.0;  ph.im = 0.0;  break;
    case 1:  ph.re = 0.0;  ph.im = -1.0; break;
    case 2:  ph.re = -1.0; ph.im = 0.0;  break;
    default: ph.re = 0.0;  ph.im = 1.0;  break;
  }
  for (int i = 0; i < 2*l + 1; ++i)
    for (int j = 0; j < 2*l + 1; ++j) q[i][j] = cmul(q[i][j], ph);
}

// out layout: [2*lo+1][2*l1+1][2*l2+1], Frobenius-normalized * sqrt(2*lo+1)
__device__ void compute_cg(int lo, int l1, int l2, float* out) {
  cplx q1[7][7], q2[7][7], qo[7][7];
  build_q(l1, q1); build_q(l2, q2); build_q(lo, qo);
  const int n1 = 2*l1 + 1, n2 = 2*l2 + 1, no = 2*lo + 1;
  double T[7][5][3];
  double nrm = 0.0;
  for (int m = 0; m < no; ++m)
    for (int j = 0; j < n1; ++j)
      for (int k = 0; k < n2; ++k) {
        double acc = 0.0;
        for (int i = 0; i < n1; ++i)
          for (int kk = 0; kk < n2; ++kk) {
            int m1 = i - l1, m2 = kk - l2, m3 = m1 + m2;
            if (m3 < -lo || m3 > lo) continue;
            double su = su2_elem(l1, l2, lo, m1, m2);
            if (su == 0.0) continue;
            cplx t = cmul(q1[i][j], q2[kk][k]);
            cplx qc = qo[lo + m3][m]; qc.im = -qc.im;   // conj
            t = cmul(t, qc);
            acc += t.re * su;
          }
        T[m][j][k] = acc;
        nrm += acc * acc;
      }
  double scale = sqrt((double)(2*lo + 1)) / sqrt(nrm);
  int idx = 0;
  for (int m = 0; m < no; ++m)
    for (int j = 0; j < n1; ++j)
      for (int k = 0; k < n2; ++k)
        out[idx++] = (float)(T[m][j][k] * scale);
}

__global__ void cg_setup_kernel(float* __restrict__ ws) {
  int i = threadIdx.x;
  if (i >= 6) return;
  const int lo[6]  = {0, 1, 2, 1, 2, 3};
  const int l1[6]  = {1, 1, 1, 2, 2, 2};
  const int off[6] = {CG_0_11, CG_1_11, CG_2_11, CG_1_21, CG_2_21, CG_3_21};
  compute_cg(lo[i], l1[i], 1, ws + off[i]);
}

// ============================ launch =======================================

extern "C" void kernel_launch(void* const* d_in, const int* in_sizes, int n_in,
                              void* d_out, int out_size, void* d_ws,
                              size_t ws_size, hipStream_t stream) {
  const float* f1 = (const float*)d_in[0];
  const float* f2 = (const float*)d_in[1];
  float* out = (float*)d_out;
  float* ws  = (float*)d_ws;     // 306 floats of CG coefficients

  const int batch  = in_sizes[0] / ROWLEN;   // 32768
  const int blocks = batch / RPB;            // 8192

  cg_setup_kernel<<<1, 64, 0, stream>>>(ws);
  etp_kernel<<<blocks, NTHREADS, 0, stream>>>(f1, f2, ws, out);
}